// LRENet_2_34342558499512
// MI455X (gfx1250) — compile-verified
//
#include <hip/hip_runtime.h>
#include <hip/hip_bf16.h>
#include <math.h>

typedef float v2f __attribute__((ext_vector_type(2)));
typedef float v8f __attribute__((ext_vector_type(8)));

#define L_    4
#define AD_   512
#define D_    512
#define T_    64
#define NTOK_ 4096
#define PROJ_ 256
#define HID_  2048
#define CAP_  16

// ---------------------------------------------------------------------------
// Generic f32 WMMA GEMM:  C = act(A @ B (+bias)) (+R)
//   A: (M,K) row-major. B: BT ? (N,K) row-major (i.e. W, use W[n,k])
//                          : (K,N) row-major.
//   TO: store C as (N,M) i.e. C[n*M+m]  (used to produce head-major K/V)
//   ACT: 0 none, 1 exact GELU, 2 x*sigmoid(1.702x)
//   blockIdx.z batch with strides sA/sB/sBias/sC.
// One wave per block; 16x64 tile = 4x V_WMMA_F32_16X16X4_F32 per K-step with
// the A fragment reused 4x. K-loop is software-pipelined: fragments for step
// k+4 are issued before the WMMA group of step k, so loads are in flight
// during matrix math (requires K >= 8 and K % 4 == 0 -- true for all calls).
// ---------------------------------------------------------------------------
template <int BT, int TO, int ACT>
__global__ __launch_bounds__(32)
void gemm_wmma_f32(const float* __restrict__ A, const float* __restrict__ B,
                   const float* __restrict__ bias, const float* __restrict__ R,
                   float* __restrict__ C,
                   int M, int N, int K,
                   long long sA, long long sB, long long sBias, long long sC)
{
    const int z = blockIdx.z;
    A += (size_t)z * sA;
    B += (size_t)z * sB;
    C += (size_t)z * sC;
    const float* bp = bias ? (bias + (size_t)z * sBias) : nullptr;

    const int lane = threadIdx.x & 31;
    const int half = lane >> 4;   // A: 0 -> K{0,1}, 1 -> K{2,3};  C: rows +0..7 vs +8..15
    const int l16  = lane & 15;

    const int tileM = blockIdx.y * 16;
    const int tileN = blockIdx.x * 64;

    v8f acc[4] = {};

    // A 16x4 fragment: lanes 0-15 hold M=0..15 {K,K+1}; lanes 16-31 {K+2,K+3}
    const float* arow = A + (size_t)(tileM + l16) * K + 2 * half;

    // Per-variant B base pointer
    const float* brow0;  // BT=1: row (tileN+l16) of W;   BT=0: row (2*half) of B
    if (BT) brow0 = B + (size_t)(tileN + l16) * K + 2 * half;
    else    brow0 = B + (size_t)(2 * half) * N + tileN + l16;

    auto loadB = [&](int k0, v2f (&bf)[4]) {
        if (BT) {
            const float* p0 = brow0 + k0;
#pragma unroll
            for (int nt = 0; nt < 4; ++nt)
                bf[nt] = *(const v2f*)(p0 + (size_t)nt * 16 * K);
        } else {
            const float* p0 = brow0 + (size_t)k0 * N;
#pragma unroll
            for (int nt = 0; nt < 4; ++nt) {
                bf[nt].x = p0[nt * 16];
                bf[nt].y = p0[(size_t)N + nt * 16];
            }
        }
    };

    // ---- software-pipelined K loop --------------------------------------
    v2f a_cur = *(const v2f*)(arow);
    v2f b_cur[4];
    loadB(0, b_cur);

#pragma unroll 4
    for (int k0 = 0; k0 < K - 4; k0 += 4) {
        // issue next-step fragments first (in flight during WMMA group)
        v2f a_nxt = *(const v2f*)(arow + k0 + 4);
        v2f b_nxt[4];
        loadB(k0 + 4, b_nxt);
#pragma unroll
        for (int nt = 0; nt < 4; ++nt)
            acc[nt] = __builtin_amdgcn_wmma_f32_16x16x4_f32(
                false, a_cur, false, b_cur[nt], (short)0, acc[nt], false, false);
        a_cur = a_nxt;
#pragma unroll
        for (int nt = 0; nt < 4; ++nt) b_cur[nt] = b_nxt[nt];
    }
    // peeled last step
#pragma unroll
    for (int nt = 0; nt < 4; ++nt)
        acc[nt] = __builtin_amdgcn_wmma_f32_16x16x4_f32(
            false, a_cur, false, b_cur[nt], (short)0, acc[nt], false, false);

    // ---- epilogue: C/D layout: VGPR r -> M = r + 8*half, N = l16 ---------
#pragma unroll
    for (int nt = 0; nt < 4; ++nt) {
        const int col = tileN + nt * 16 + l16;
        const float bv = bp ? bp[col] : 0.0f;
#pragma unroll
        for (int r = 0; r < 8; ++r) {
            const int row = tileM + r + 8 * half;
            float v = acc[nt][r] + bv;
            if (ACT == 1)      v = 0.5f * v * (1.0f + erff(v * 0.70710678118654752f));
            else if (ACT == 2) v = v / (1.0f + __expf(-1.702f * v));
            if (R) v += R[(size_t)row * N + col];
            if (TO) C[(size_t)col * M + row] = v;
            else    C[(size_t)row * N + col] = v;
        }
    }
}

// ---------------------------------------------------------------------------
// Row LayerNorm (variance ddof=0, eps 1e-5). One block per row, 256 threads.
// ---------------------------------------------------------------------------
__global__ void layernorm_kernel(const float* __restrict__ x, int ldx,
                                 const float* __restrict__ g, const float* __restrict__ b,
                                 float* __restrict__ y, int C)
{
    __shared__ float s1[256], s2[256];
    const int row = blockIdx.x;
    const float* xr = x + (size_t)row * ldx;
    float a = 0.f, q = 0.f;
    for (int c = threadIdx.x; c < C; c += 256) { float v = xr[c]; a += v; q += v * v; }
    s1[threadIdx.x] = a; s2[threadIdx.x] = q; __syncthreads();
    for (int st = 128; st > 0; st >>= 1) {
        if (threadIdx.x < st) { s1[threadIdx.x] += s1[threadIdx.x + st]; s2[threadIdx.x] += s2[threadIdx.x + st]; }
        __syncthreads();
    }
    const float mu  = s1[0] / (float)C;
    const float var = s2[0] / (float)C - mu * mu;
    const float inv = rsqrtf(var + 1e-5f);
    for (int c = threadIdx.x; c < C; c += 256)
        y[(size_t)row * C + c] = (xr[c] - mu) * inv * g[c] + b[c];
}

// ---------------------------------------------------------------------------
// Per-layer cross attention with hd = 1 (512 heads of dim 1).
// scores[h][t][s] = q[t,h]*k[s,h]  (scale = 1/sqrt(1) = 1)
// Row max over s is q*kmax (q>=0) or q*kmin (q<0)  -> single pass softmax.
// One block per head; K/V columns are head-major (KT[h][s]) from transOut GEMM.
// ---------------------------------------------------------------------------
__global__ void attn_ind_kernel(const float* __restrict__ Q,
                                const float* __restrict__ KT,
                                const float* __restrict__ VT,
                                float* __restrict__ Out)
{
    __shared__ float ks[NTOK_], vs[NTOK_];
    __shared__ float qs[T_];
    __shared__ float red[256];
    __shared__ float dsh[256], nsh[256];
    __shared__ float kmaxS, kminS;
    const int h = blockIdx.x, tid = threadIdx.x;

    for (int i = tid; i < NTOK_; i += 256) {
        ks[i] = KT[(size_t)h * NTOK_ + i];
        vs[i] = VT[(size_t)h * NTOK_ + i];
    }
    if (tid < T_) qs[tid] = Q[(size_t)tid * AD_ + h];
    __syncthreads();

    float mx = -1e30f, mn = 1e30f;
    for (int i = tid; i < NTOK_; i += 256) { float k = ks[i]; mx = fmaxf(mx, k); mn = fminf(mn, k); }
    red[tid] = mx; __syncthreads();
    for (int st = 128; st > 0; st >>= 1) { if (tid < st) red[tid] = fmaxf(red[tid], red[tid + st]); __syncthreads(); }
    if (tid == 0) kmaxS = red[0]; __syncthreads();
    red[tid] = mn; __syncthreads();
    for (int st = 128; st > 0; st >>= 1) { if (tid < st) red[tid] = fminf(red[tid], red[tid + st]); __syncthreads(); }
    if (tid == 0) kminS = red[0]; __syncthreads();

    const int t = tid & 63, part = tid >> 6;       // 64 tokens x 4 s-segments
    const float qv = qs[t];
    const float m  = (qv >= 0.f) ? qv * kmaxS : qv * kminS;
    float den = 0.f, num = 0.f;
    const int s0 = part * (NTOK_ / 4);
    for (int s = s0; s < s0 + NTOK_ / 4; ++s) {
        float e = __expf(qv * ks[s] - m);
        den += e; num += e * vs[s];
    }
    dsh[tid] = den; nsh[tid] = num; __syncthreads();
    if (tid < T_) {
        float dd = dsh[tid] + dsh[tid + 64] + dsh[tid + 128] + dsh[tid + 192];
        float nn = nsh[tid] + nsh[tid + 64] + nsh[tid + 128] + nsh[tid + 192];
        Out[(size_t)tid * AD_ + h] = nn / dd;
    }
}

// ---------------------------------------------------------------------------
// GSA self-attention: 8 heads, hd = 64, T = 64. One block per head, all in LDS.
// QKV layout: [t][0:512]=q, [512:1024]=k, [1024:1536]=v, head h -> cols h*64..
// ---------------------------------------------------------------------------
__global__ void gsa_attn_kernel(const float* __restrict__ QKV, float* __restrict__ Out)
{
    __shared__ float k[64][65], v[64][65], sc[64][65];
    const int h = blockIdx.x, tid = threadIdx.x;
    for (int i = tid; i < 4096; i += 256) {
        int t = i >> 6, d = i & 63;
        k[t][d] = QKV[(size_t)t * 1536 + 512  + h * 64 + d];
        v[t][d] = QKV[(size_t)t * 1536 + 1024 + h * 64 + d];
    }
    __syncthreads();
    for (int i = tid; i < 4096; i += 256) {
        int t = i >> 6, s = i & 63;
        const float* qr = QKV + (size_t)t * 1536 + h * 64;
        float acc = 0.f;
        for (int d = 0; d < 64; ++d) acc += qr[d] * k[s][d];
        sc[t][s] = acc * 0.125f;   // 1/sqrt(64)
    }
    __syncthreads();
    if (tid < 64) {
        float mx = -1e30f;
        for (int s = 0; s < 64; ++s) mx = fmaxf(mx, sc[tid][s]);
        float sum = 0.f;
        for (int s = 0; s < 64; ++s) { float e = __expf(sc[tid][s] - mx); sc[tid][s] = e; sum += e; }
        float inv = 1.f / sum;
        for (int s = 0; s < 64; ++s) sc[tid][s] *= inv;
    }
    __syncthreads();
    for (int i = tid; i < 4096; i += 256) {
        int t = i >> 6, d = i & 63;
        float acc = 0.f;
        for (int s = 0; s < 64; ++s) acc += sc[t][s] * v[s][d];
        Out[(size_t)t * D_ + h * 64 + d] = acc;
    }
}

// ---------------------------------------------------------------------------
// MoE top-1 routing on 64 tokens / 4 experts / capacity 16. Single block.
// Stable descending sort by gate via rank counting (matches jnp.argsort(-gate)).
// ---------------------------------------------------------------------------
__global__ void moe_route_kernel(const float* __restrict__ qf, const float* __restrict__ P,
                                 const float* __restrict__ sim, const float* __restrict__ temp,
                                 float* __restrict__ buf,
                                 int* __restrict__ expertArr, int* __restrict__ posArr,
                                 float* __restrict__ gateArr, float* __restrict__ auxAcc)
{
    __shared__ float sn[4], pn[64];
    __shared__ float lg[64][4], scs[64][4];
    __shared__ int   es[64];
    __shared__ float gs[64];
    __shared__ int   rk[64], ps[64], kp[64];
    const int tid = threadIdx.x;

    for (int i = tid; i < L_ * CAP_ * D_; i += 256) buf[i] = 0.f;
    if (tid < 4) {
        float s = 0.f;
        for (int k2 = 0; k2 < PROJ_; ++k2) { float v = sim[k2 * 4 + tid]; s += v * v; }
        sn[tid] = sqrtf(s) + 1e-8f;
    }
    if (tid < 64) {
        float s = 0.f; const float* pr = P + (size_t)tid * PROJ_;
        for (int k2 = 0; k2 < PROJ_; ++k2) { float v = pr[k2]; s += v * v; }
        pn[tid] = sqrtf(s) + 1e-8f;
    }
    __syncthreads();
    const float escale = __expf(fminf(temp[0], logf(100.f)));
    {
        const int t = tid >> 2, e = tid & 3;
        const float* pr = P + (size_t)t * PROJ_;
        float dot = 0.f;
        for (int k2 = 0; k2 < PROJ_; ++k2) dot += pr[k2] * sim[k2 * 4 + e];
        lg[t][e] = dot / (pn[t] * sn[e]) * escale;
    }
    __syncthreads();
    if (tid < 64) {
        float mx = lg[tid][0]; int be = 0;
        for (int e = 1; e < 4; ++e) if (lg[tid][e] > mx) { mx = lg[tid][e]; be = e; }
        float sum = 0.f, ex[4];
        for (int e = 0; e < 4; ++e) { ex[e] = __expf(lg[tid][e] - mx); sum += ex[e]; }
        for (int e = 0; e < 4; ++e) scs[tid][e] = ex[e] / sum;
        es[tid] = be; gs[tid] = scs[tid][be];
    }
    __syncthreads();
    if (tid < 64) {
        const float g = gs[tid]; int cnt = 0;
        for (int u = 0; u < 64; ++u) { float gu = gs[u]; if (gu > g || (gu == g && u < tid)) cnt++; }
        rk[tid] = cnt;
    }
    __syncthreads();
    if (tid < 64) {
        const int e = es[tid], r = rk[tid]; int cnt = 0;
        for (int u = 0; u < 64; ++u) if (es[u] == e && rk[u] < r) cnt++;
        const int keep = (cnt < CAP_);
        kp[tid] = keep;
        ps[tid] = keep ? cnt : 0;
        expertArr[tid] = e;
        posArr[tid]    = ps[tid];
        gateArr[tid]   = keep ? gs[tid] : 0.f;
    }
    __syncthreads();
    for (int t2 = 0; t2 < 64; ++t2) {
        if (kp[t2]) {
            float* dst = buf + ((size_t)es[t2] * CAP_ + ps[t2]) * D_;
            const float* src = qf + (size_t)t2 * D_;
            for (int c = tid; c < D_; c += 256) dst[c] = src[c];
        }
    }
    if (tid == 0) {
        float la = 0.f;
        for (int e = 0; e < 4; ++e) {
            float me = 0.f; int cc = 0;
            for (int t2 = 0; t2 < 64; ++t2) { me += scs[t2][e]; cc += (es[t2] == e); }
            la += (me / 64.f) * ((float)cc / 64.f);
        }
        auxAcc[0] += la * 4.f * 0.01f;   // * E * AUX_W
    }
}

__global__ void moe_gather_kernel(const float* __restrict__ Y,
                                  const int* __restrict__ expertArr, const int* __restrict__ posArr,
                                  const float* __restrict__ gateArr,
                                  float* __restrict__ cur, float* __restrict__ feat1)
{
    const int t = blockIdx.x;
    const int e = expertArr[t], p = posArr[t];
    const float gk = gateArr[t];
    const float* src = Y + ((size_t)e * CAP_ + p) * D_;
    for (int c = threadIdx.x; c < D_; c += blockDim.x) {
        float v = cur[(size_t)t * D_ + c] + src[c] * gk;
        cur[(size_t)t * D_ + c]   = v;
        feat1[(size_t)t * D_ + c] = v;
    }
}

// ---------------------------------------------------------------------------
// Final classifier head: logits = hb_flat(16384) @ cls_w.T + cls_b, hazards,
// Y_hat = argmax, aux copy. Single block.
// ---------------------------------------------------------------------------
__global__ void head_kernel(const float* __restrict__ hb, const float* __restrict__ clsw,
                            const float* __restrict__ clsb, const float* __restrict__ auxAcc,
                            float* __restrict__ out)
{
    __shared__ float r0[256], r1[256];
    const int tid = threadIdx.x;
    float a0 = 0.f, a1 = 0.f;
    for (int j = tid; j < T_ * 256; j += 256) {
        float h = hb[j];
        a0 += h * clsw[j];
        a1 += h * clsw[T_ * 256 + j];
    }
    r0[tid] = a0; r1[tid] = a1; __syncthreads();
    for (int st = 128; st > 0; st >>= 1) {
        if (tid < st) { r0[tid] += r0[tid + st]; r1[tid] += r1[tid + st]; }
        __syncthreads();
    }
    if (tid == 0) {
        float l0 = r0[0] + clsb[0], l1 = r1[0] + clsb[1];
        out[0] = l0;
        out[1] = l1;
        out[2] = 1.f / (1.f + __expf(-l0));
        out[3] = 1.f / (1.f + __expf(-l1));
        out[4] = (l1 > l0) ? 1.f : 0.f;   // argmax, first index on ties
        out[5] = auxAcc[0];
    }
}

__global__ void zero_kernel(float* p, int n)
{ int i = blockIdx.x * 256 + threadIdx.x; if (i < n) p[i] = 0.f; }

__global__ void scale_kernel(float* dst, const float* src, int n, float s)
{ int i = blockIdx.x * 256 + threadIdx.x; if (i < n) dst[i] = src[i] * s; }

// ---------------------------------------------------------------------------
extern "C" void kernel_launch(void* const* d_in, const int* in_sizes, int n_in,
                              void* d_out, int out_size, void* d_ws, size_t ws_size,
                              hipStream_t stream)
{
    const float* share    = (const float*)d_in[0];
    const float* fet      = (const float*)d_in[1];
    const float* ln1g     = (const float*)d_in[2];
    const float* ln1b     = (const float*)d_in[3];
    const float* ln2g     = (const float*)d_in[4];
    const float* ln2b     = (const float*)d_in[5];
    const float* qkvw     = (const float*)d_in[6];
    const float* qkvb     = (const float*)d_in[7];
    const float* outw     = (const float*)d_in[8];
    const float* outb     = (const float*)d_in[9];
    const float* mlpw     = (const float*)d_in[10];
    const float* mlpb     = (const float*)d_in[11];
    const float* gateproj = (const float*)d_in[12];
    const float* gatesim  = (const float*)d_in[13];
    const float* gatetemp = (const float*)d_in[14];
    const float* expw1    = (const float*)d_in[15];
    const float* expb1    = (const float*)d_in[16];
    const float* expw2    = (const float*)d_in[17];
    const float* expb2    = (const float*)d_in[18];
    const float* g1g      = (const float*)d_in[19];
    const float* g1b      = (const float*)d_in[20];
    const float* gqkvw    = (const float*)d_in[21];
    const float* gqkvb    = (const float*)d_in[22];
    const float* goutw    = (const float*)d_in[23];
    const float* goutb    = (const float*)d_in[24];
    const float* g2g      = (const float*)d_in[25];
    const float* g2b      = (const float*)d_in[26];
    const float* fcw      = (const float*)d_in[27];
    const float* fcb      = (const float*)d_in[28];
    const float* projw    = (const float*)d_in[29];
    const float* projb    = (const float*)d_in[30];
    const float* butterw  = (const float*)d_in[31];
    const float* butterb  = (const float*)d_in[32];
    const float* clsw     = (const float*)d_in[33];
    const float* clsb     = (const float*)d_in[34];

    float* out = (float*)d_out;
    float* ws  = (float*)d_ws;

    size_t off = 0;
    auto alloc = [&](size_t n) { float* p = ws + off; off += n; return p; };

    float* sfln  = alloc((size_t)NTOK_ * AD_);     // 2M
    float* KT    = alloc((size_t)1024 * NTOK_);    // 4M (rows 0..511 = K heads, 512..1023 = V heads)
    float* tokln = alloc((size_t)T_ * AD_);
    float* Qb    = alloc((size_t)T_ * AD_);
    float* attnO = alloc((size_t)T_ * AD_);
    float* attnP = alloc((size_t)T_ * AD_);
    float* qf    = alloc((size_t)T_ * D_);
    float* Pm    = alloc((size_t)T_ * PROJ_);
    float* buf   = alloc((size_t)L_ * CAP_ * D_);
    float* hbuf  = alloc((size_t)L_ * CAP_ * HID_);
    float* ybuf  = alloc((size_t)L_ * CAP_ * D_);
    float* cur   = alloc((size_t)T_ * D_);
    float* curS  = alloc((size_t)T_ * D_);
    float* x1    = alloc((size_t)T_ * D_);
    float* qkv   = alloc((size_t)T_ * 3 * D_);
    float* attnG = alloc((size_t)T_ * D_);
    float* xb    = alloc((size_t)T_ * D_);
    float* x2    = alloc((size_t)T_ * D_);
    float* hfc   = alloc((size_t)T_ * HID_);
    float* gated = alloc((size_t)T_ * D_);
    float* hbb   = alloc((size_t)T_ * 256);
    float* auxA  = alloc(64);                       // [0] = aux accumulator
    int*   eArr  = (int*)alloc(64);
    int*   pArr  = (int*)alloc(64);
    float* gArr  = alloc(64);

    const int TD = T_ * D_;                         // 32768

    // ---- init: cur = 0, aux = 0, feature2 + feature2_pre = 0 -------------
    zero_kernel<<<(TD + 255) / 256, 256, 0, stream>>>(cur, TD);
    zero_kernel<<<1, 256, 0, stream>>>(auxA, 1);
    {
        const int n2 = L_ * TD + L_ * 2;            // feature2 + feature2_pre
        zero_kernel<<<(n2 + 255) / 256, 256, 0, stream>>>(out + 6 + (size_t)L_ * TD, n2);
    }

    // ---- per-layer pipeline ---------------------------------------------
    for (int i = 0; i < L_; ++i) {
        // sf = LN(share[:, i*512:(i+1)*512])
        layernorm_kernel<<<NTOK_, 256, 0, stream>>>(
            share + (size_t)i * AD_, L_ * AD_, ln1g + (size_t)i * AD_, ln1b + (size_t)i * AD_, sfln, AD_);

        // [K|V]^T (head-major, 1024 x 4096) = (sf @ Wkv.T + bkv)^T
        gemm_wmma_f32<1, 1, 0><<<dim3(1024 / 64, NTOK_ / 16, 1), 32, 0, stream>>>(
            sfln, qkvw + (size_t)i * 3 * AD_ * AD_ + (size_t)AD_ * AD_,
            qkvb + (size_t)i * 3 * AD_ + AD_, nullptr, KT,
            NTOK_, 1024, AD_, 0, 0, 0, 0);

        // tok = LN(fet_tokens[i]); Q = tok @ Wq.T + bq
        layernorm_kernel<<<T_, 256, 0, stream>>>(
            fet + (size_t)i * T_ * AD_, AD_, ln2g + (size_t)i * AD_, ln2b + (size_t)i * AD_, tokln, AD_);
        gemm_wmma_f32<1, 0, 0><<<dim3(AD_ / 64, T_ / 16, 1), 32, 0, stream>>>(
            tokln, qkvw + (size_t)i * 3 * AD_ * AD_, qkvb + (size_t)i * 3 * AD_, nullptr, Qb,
            T_, AD_, AD_, 0, 0, 0, 0);

        // 512 heads of dim 1
        attn_ind_kernel<<<AD_, 256, 0, stream>>>(Qb, KT, KT + (size_t)AD_ * NTOK_, attnO);

        // out-proj, mlp
        gemm_wmma_f32<1, 0, 0><<<dim3(AD_ / 64, T_ / 16, 1), 32, 0, stream>>>(
            attnO, outw + (size_t)i * AD_ * AD_, outb + (size_t)i * AD_, nullptr, attnP,
            T_, AD_, AD_, 0, 0, 0, 0);
        gemm_wmma_f32<1, 0, 0><<<dim3(D_ / 64, T_ / 16, 1), 32, 0, stream>>>(
            attnP, mlpw + (size_t)i * D_ * AD_, mlpb + (size_t)i * D_, nullptr, qf,
            T_, D_, AD_, 0, 0, 0, 0);

        // MoE: p = qf @ gate_proj_w (K x N layout)
        gemm_wmma_f32<0, 0, 0><<<dim3(PROJ_ / 64, T_ / 16, 1), 32, 0, stream>>>(
            qf, gateproj, nullptr, nullptr, Pm, T_, PROJ_, D_, 0, 0, 0, 0);
        moe_route_kernel<<<1, 256, 0, stream>>>(qf, Pm, gatesim, gatetemp, buf, eArr, pArr, gArr, auxA);

        // expert FFN (batched over 4 experts), GELU fused into GEMM 1
        gemm_wmma_f32<0, 0, 1><<<dim3(HID_ / 64, CAP_ / 16, L_), 32, 0, stream>>>(
            buf, expw1, expb1, nullptr, hbuf, CAP_, HID_, D_,
            (long long)CAP_ * D_, (long long)D_ * HID_, HID_, (long long)CAP_ * HID_);
        gemm_wmma_f32<0, 0, 0><<<dim3(D_ / 64, CAP_ / 16, L_), 32, 0, stream>>>(
            hbuf, expw2, expb2, nullptr, ybuf, CAP_, D_, HID_,
            (long long)CAP_ * HID_, (long long)HID_ * D_, D_, (long long)CAP_ * D_);

        // cur += gated expert output; feature1[i] = cur
        moe_gather_kernel<<<T_, 256, 0, stream>>>(ybuf, eArr, pArr, gArr, cur, out + 6 + (size_t)i * TD);
    }

    // ---- GSA block -------------------------------------------------------
    scale_kernel<<<(TD + 255) / 256, 256, 0, stream>>>(curS, cur, TD, 1.0f / L_);
    layernorm_kernel<<<T_, 256, 0, stream>>>(curS, D_, g1g, g1b, x1, D_);
    gemm_wmma_f32<1, 0, 0><<<dim3(3 * D_ / 64, T_ / 16, 1), 32, 0, stream>>>(
        x1, gqkvw, gqkvb, nullptr, qkv, T_, 3 * D_, D_, 0, 0, 0, 0);
    gsa_attn_kernel<<<D_ / 64, 256, 0, stream>>>(qkv, attnG);
    gemm_wmma_f32<1, 0, 0><<<dim3(D_ / 64, T_ / 16, 1), 32, 0, stream>>>(
        attnG, goutw, goutb, curS, xb, T_, D_, D_, 0, 0, 0, 0);
    layernorm_kernel<<<T_, 256, 0, stream>>>(xb, D_, g2g, g2b, x2, D_);
    gemm_wmma_f32<1, 0, 2><<<dim3(HID_ / 64, T_ / 16, 1), 32, 0, stream>>>(
        x2, fcw, fcb, nullptr, hfc, T_, HID_, D_, 0, 0, 0, 0);   // x*sigmoid(1.702x)
    gemm_wmma_f32<1, 0, 0><<<dim3(D_ / 64, T_ / 16, 1), 32, 0, stream>>>(
        hfc, projw, projb, xb, gated, T_, D_, HID_, 0, 0, 0, 0); // + residual x
    gemm_wmma_f32<1, 0, 0><<<dim3(256 / 64, T_ / 16, 1), 32, 0, stream>>>(
        gated, butterw, butterb, nullptr, hbb, T_, 256, D_, 0, 0, 0, 0);

    // ---- head: logits / hazards / Y_hat / aux ---------------------------
    head_kernel<<<1, 256, 0, stream>>>(hbb, clsw, clsb, auxA, out);
}